// AdaptiveTripletMarginLoss_57380763074766
// MI455X (gfx1250) — compile-verified
//
#include <hip/hip_runtime.h>
#include <math.h>

// CDNA5 / gfx1250: wave32, WMMA f32 16x16x4, async global->LDS (ASYNCcnt).
typedef __attribute__((ext_vector_type(2))) float v2f;
typedef __attribute__((ext_vector_type(8))) float v8f;

#define DDIM 256
#define CHUNK 32
#define NCHUNK (DDIM / CHUNK)          // 8 column chunks
#define PAD 36                          // tile row stride in floats (144B: 16B-aligned, conflict-free reads)
#define NWAVES 4
#define THREADS (NWAVES * 32)
#define ROWS_PER_WAVE 16
#define ROWS_PER_BLOCK (NWAVES * ROWS_PER_WAVE)   // 64
#define TILE_FLTS (ROWS_PER_WAVE * PAD)           // 576
#define TILE_B (TILE_FLTS * 4)                    // 2304 bytes (16B multiple)
#define BUF_B (3 * TILE_B)                        // 6912 bytes (16B multiple)

// Each wave owns 16 rows and a private double-buffered LDS tile set (a/p/n).
// Chunk ch+1 is DMA'd memory->LDS with global_load_async_to_lds_b128 while
// chunk ch is consumed by V_WMMA_F32_16X16X4_F32 (ones-B => per-row sum of squares).
__global__ __launch_bounds__(THREADS) void triplet_main_kernel(
    const float* __restrict__ A, const float* __restrict__ P,
    const float* __restrict__ Nn, float* __restrict__ partials)
{
    __shared__ __align__(16) float lds[NWAVES][2][3][TILE_FLTS];   // 54 KB
    __shared__ float wsum[NWAVES];

    const int tid  = threadIdx.x;
    const int lane = tid & 31;
    const int wave = tid >> 5;
    const int row0 = blockIdx.x * ROWS_PER_BLOCK + wave * ROWS_PER_WAVE;

    float* tile0 = &lds[wave][0][0][0];
    // LDS aperture lives in flat-address bits [63:32]; low 32 bits are the LDS byte offset.
    const unsigned lds0 = (unsigned)(uintptr_t)tile0;

    v8f c_ap = {0.f,0.f,0.f,0.f,0.f,0.f,0.f,0.f};
    v8f c_an = {0.f,0.f,0.f,0.f,0.f,0.f,0.f,0.f};
    v8f c_pn = {0.f,0.f,0.f,0.f,0.f,0.f,0.f,0.f};
    const v2f bones = {1.0f, 1.0f};

    // async-DMA lane mapping: lane -> (row = lane/8 + 4i, 16B column slot = lane%8)
    const int lrow = lane >> 3;
    const int lcol = (lane & 7) << 2;        // float offset within chunk
    // WMMA A-fragment mapping: M = lane%16, K = 2*(lane/16) + {0,1}
    const int m    = lane & 15;
    const int koff = (lane >> 4) << 1;

    // Issue 12 async b128 DMAs (a/p/n, 16x32 tile) into buffer `buf` for chunk `ch`.
    auto prefetch = [&](int buf, int ch) {
#pragma unroll
        for (int i = 0; i < 4; ++i) {
            const int r = lrow + (i << 2);                         // 0..15
            const int g = (row0 + r) * DDIM + ch * CHUNK + lcol;   // global float index
            const unsigned loff = lds0 + (unsigned)buf * BUF_B + (unsigned)((r * PAD + lcol) * 4);
            asm volatile("global_load_async_to_lds_b128 %0, %1, off"
                         :: "v"(loff),                "v"((const void*)(A  + g)) : "memory");
            asm volatile("global_load_async_to_lds_b128 %0, %1, off"
                         :: "v"(loff + TILE_B),       "v"((const void*)(P  + g)) : "memory");
            asm volatile("global_load_async_to_lds_b128 %0, %1, off"
                         :: "v"(loff + 2u * TILE_B),  "v"((const void*)(Nn + g)) : "memory");
        }
    };

    // Consume buffer `buf`: 8 K-steps of 3 WMMAs on squared diffs.
    auto compute = [&](int buf) {
        const float* ta = tile0 + (buf * 3 + 0) * TILE_FLTS;
        const float* tp = tile0 + (buf * 3 + 1) * TILE_FLTS;
        const float* tn = tile0 + (buf * 3 + 2) * TILE_FLTS;
#pragma unroll
        for (int kk = 0; kk < CHUNK / 4; ++kk) {
            const int o = m * PAD + (kk << 2) + koff;              // even -> 8B-aligned b64
            const float2 a2 = *reinterpret_cast<const float2*>(ta + o);
            const float2 p2 = *reinterpret_cast<const float2*>(tp + o);
            const float2 n2 = *reinterpret_cast<const float2*>(tn + o);
            v2f dap = {a2.x - p2.x, a2.y - p2.y};
            v2f dan = {a2.x - n2.x, a2.y - n2.y};
            v2f dpn = {p2.x - n2.x, p2.y - n2.y};
            dap *= dap; dan *= dan; dpn *= dpn;
            c_ap = __builtin_amdgcn_wmma_f32_16x16x4_f32(false, dap, false, bones,
                                                         (short)0, c_ap, false, false);
            c_an = __builtin_amdgcn_wmma_f32_16x16x4_f32(false, dan, false, bones,
                                                         (short)0, c_an, false, false);
            c_pn = __builtin_amdgcn_wmma_f32_16x16x4_f32(false, dpn, false, bones,
                                                         (short)0, c_pn, false, false);
        }
    };

    prefetch(0, 0);
#pragma unroll
    for (int ch = 0; ch < NCHUNK; ++ch) {
        if (ch + 1 < NCHUNK) {
            prefetch((ch + 1) & 1, ch + 1);
            // async loads complete in order: <=12 outstanding => this chunk's 12 are done.
            asm volatile("s_wait_asynccnt 12" ::: "memory");
        } else {
            asm volatile("s_wait_asynccnt 0" ::: "memory");
        }
        compute(ch & 1);
        // Drain LDS reads before this buffer is overwritten next iteration (WAR).
        asm volatile("s_wait_dscnt 0" ::: "memory");
    }

    // C layout: VGPR j, lanes 0-15 -> row j; lanes 16-31 -> row 8+j (all columns equal).
    float acc = 0.0f;
#pragma unroll
    for (int j = 0; j < 8; ++j) {
        const float dap = sqrtf(c_ap[j]);
        const float dan = sqrtf(c_an[j]);
        const float dpn = sqrtf(c_pn[j]);
        const float ms = 1.0f + 2.0f / (expf(4.0f * dap) + 1e-6f);
        const float md = 1.0f + 2.0f / (expf(4.0f - 4.0f * dan) + 1e-6f);
        acc += dap - 0.5f * (dan + dpn) + ms + md;
    }
    // lanes 0-15 hold rows 0..7 sum; lanes 16-31 hold rows 8..15 sum.
    const float wtot = __shfl(acc, 0, 32) + __shfl(acc, 16, 32);
    if (lane == 0) wsum[wave] = wtot;
    __syncthreads();
    if (tid == 0) {
        float b = 0.0f;
#pragma unroll
        for (int w = 0; w < NWAVES; ++w) b += wsum[w];
        partials[blockIdx.x] = b;
    }
}

// Deterministic final reduction of block partials -> mean.
__global__ __launch_bounds__(256) void triplet_finish_kernel(
    const float* __restrict__ partials, int n, float invB, float* __restrict__ out)
{
    __shared__ float s[256];
    float acc = 0.0f;
    for (int i = threadIdx.x; i < n; i += 256) acc += partials[i];
    s[threadIdx.x] = acc;
    __syncthreads();
    for (int off = 128; off > 0; off >>= 1) {
        if ((int)threadIdx.x < off) s[threadIdx.x] += s[threadIdx.x + off];
        __syncthreads();
    }
    if (threadIdx.x == 0) out[0] = s[0] * invB;
}

extern "C" void kernel_launch(void* const* d_in, const int* in_sizes, int n_in,
                              void* d_out, int out_size, void* d_ws, size_t ws_size,
                              hipStream_t stream) {
    (void)n_in; (void)out_size; (void)ws_size;
    const float* A  = (const float*)d_in[0];
    const float* P  = (const float*)d_in[1];
    const float* Nn = (const float*)d_in[2];
    float* out      = (float*)d_out;
    float* partials = (float*)d_ws;

    const int B = in_sizes[0] / DDIM;          // 65536
    const int nblocks = B / ROWS_PER_BLOCK;    // 1024

    triplet_main_kernel<<<nblocks, THREADS, 0, stream>>>(A, P, Nn, partials);
    triplet_finish_kernel<<<1, 256, 0, stream>>>(partials, nblocks, 1.0f / (float)B, out);
}